// DensityGraphNet_40132174414349
// MI455X (gfx1250) — compile-verified
//
#include <hip/hip_runtime.h>
#include <hip/hip_bf16.h>
#include <math.h>

// ---------------------------------------------------------------------------
// CDNA5 (gfx1250) message-passing GNN layer.
// GEMMs use V_WMMA_F32_16X16X4_F32 (exact fp32 tensor op, wave32).
// One wave owns one 16-row tile. Weights are pre-packed per launch into
// WMMA fragment order so each B fragment is ONE global_load_b64.
// ---------------------------------------------------------------------------

typedef float v2f __attribute__((ext_vector_type(2)));
typedef float v8f __attribute__((ext_vector_type(8)));

#define LN_EPS 1e-5f

__device__ __forceinline__ v8f wmma4(v2f a, v2f b, v8f c) {
  // D(16x16,f32) += A(16x4,f32) * B(4x16,f32)
  return __builtin_amdgcn_wmma_f32_16x16x4_f32(
      /*neg_a=*/false, a, /*neg_b=*/false, b,
      /*c_mod=*/(short)0, c, /*reuse_a=*/false, /*reuse_b=*/false);
}

// ---------------------------------------------------------------------------
// Weight pre-pack: W[KIN][128] row-major -> fragment order:
//   Wp[((K*8 + n)*32 + lane)] = { W[4K+2hf][16n+col], W[4K+2hf+1][16n+col] }
// so a lane's B fragment for (K-step, col-tile n) is one 8-byte load.
// ---------------------------------------------------------------------------
__global__ void pack_w_kernel(const float* __restrict__ W, float* __restrict__ Wp,
                              int KIN) {
  const int idx = blockIdx.x * blockDim.x + threadIdx.x;
  const int total = (KIN / 4) * 8 * 32;      // fragments * lanes
  if (idx >= total) return;
  const int lane = idx & 31;
  const int n    = (idx >> 5) & 7;
  const int K    = idx >> 8;
  const int hf   = lane >> 4;
  const int col  = lane & 15;
  const int r0   = 4 * K + 2 * hf;
  Wp[(size_t)idx * 2 + 0] = W[(size_t)r0 * 128 + n * 16 + col];
  Wp[(size_t)idx * 2 + 1] = W[(size_t)(r0 + 1) * 128 + n * 16 + col];
}

// ---------------------------------------------------------------------------
// Layer 1:  H = SiLU(LayerNorm(A @ W1 + b1))
//   Alds: [16][KIN] (LDS),  Wp: packed fragments,  Hlds: [16][128] (LDS)
// ---------------------------------------------------------------------------
template<int KIN>
__device__ __forceinline__ void layer1_ln_silu(
    const float* Alds, const v2f* __restrict__ Wp,
    const float* __restrict__ b1, const float* __restrict__ lng,
    const float* __restrict__ lnb, float* Hlds) {
  const int lane = threadIdx.x & 31;
  const int hf   = lane >> 4;
  const int col  = lane & 15;

  v8f acc[8];
#pragma unroll
  for (int n = 0; n < 8; ++n) acc[n] = (v8f){0.f,0.f,0.f,0.f,0.f,0.f,0.f,0.f};

  for (int K = 0; K < KIN / 4; ++K) {
    v2f a;
    a.x = Alds[col * KIN + 4 * K + 2 * hf];
    a.y = Alds[col * KIN + 4 * K + 2 * hf + 1];
    const v2f* wk = Wp + (size_t)K * 256 + lane;   // 8 tiles * 32 lanes
#pragma unroll
    for (int n = 0; n < 8; ++n) {
      v2f b = wk[n * 32];                          // one global_load_b64
      acc[n] = wmma4(a, b, acc[n]);
    }
  }

  float b1c[8], gc[8], bc[8];
#pragma unroll
  for (int n = 0; n < 8; ++n) {
    b1c[n] = b1[n * 16 + col];
    gc[n]  = lng[n * 16 + col];
    bc[n]  = lnb[n * 16 + col];
  }

  // LayerNorm per row: acc[n][v] = elem(row = v+8*hf, col = 16n+col).
  // xor masks 1,2,4,8 reduce across the 16 lanes of this half.
#pragma unroll
  for (int v = 0; v < 8; ++v) {
    float s = 0.f;
#pragma unroll
    for (int n = 0; n < 8; ++n) { acc[n][v] += b1c[n]; s += acc[n][v]; }
    s += __shfl_xor(s, 1); s += __shfl_xor(s, 2);
    s += __shfl_xor(s, 4); s += __shfl_xor(s, 8);
    const float mean = s * (1.f / 128.f);

    float q = 0.f;
#pragma unroll
    for (int n = 0; n < 8; ++n) { float d = acc[n][v] - mean; q += d * d; }
    q += __shfl_xor(q, 1); q += __shfl_xor(q, 2);
    q += __shfl_xor(q, 4); q += __shfl_xor(q, 8);
    const float rstd = rsqrtf(q * (1.f / 128.f) + LN_EPS);

    const int row = v + 8 * hf;
#pragma unroll
    for (int n = 0; n < 8; ++n) {
      float h = (acc[n][v] - mean) * rstd * gc[n] + bc[n];
      h = h / (1.f + __expf(-h));                  // SiLU
      Hlds[row * 128 + n * 16 + col] = h;
    }
  }
}

// ---------------------------------------------------------------------------
// Layer 2:  out = H @ W2 + b2   (H: [16][128] LDS, Wp packed)
// ---------------------------------------------------------------------------
__device__ __forceinline__ void layer2_128(
    const float* Hlds, const v2f* __restrict__ Wp,
    const float* __restrict__ b2, v8f out[8]) {
  const int lane = threadIdx.x & 31;
  const int hf   = lane >> 4;
  const int col  = lane & 15;
#pragma unroll
  for (int n = 0; n < 8; ++n) out[n] = (v8f){0.f,0.f,0.f,0.f,0.f,0.f,0.f,0.f};
  for (int K = 0; K < 32; ++K) {
    v2f a;
    a.x = Hlds[col * 128 + 4 * K + 2 * hf];
    a.y = Hlds[col * 128 + 4 * K + 2 * hf + 1];
    const v2f* wk = Wp + (size_t)K * 256 + lane;
#pragma unroll
    for (int n = 0; n < 8; ++n) {
      v2f b = wk[n * 32];
      out[n] = wmma4(a, b, out[n]);
    }
  }
#pragma unroll
  for (int n = 0; n < 8; ++n) {
    const float bb = b2[n * 16 + col];
#pragma unroll
    for (int v = 0; v < 8; ++v) out[n][v] += bb;
  }
}

// ---------------------------------------------------------------------------
// Kernel 0: zero the aggregation workspace
// ---------------------------------------------------------------------------
__global__ void zero_f32(float* __restrict__ p, long n) {
  long i = (long)blockIdx.x * blockDim.x + threadIdx.x;
  if (i < n) p[i] = 0.f;
}

// ---------------------------------------------------------------------------
// Kernel 1: per-edge message + gate MLPs, scatter-add gate*msg into aggr[dst]
// ---------------------------------------------------------------------------
__global__ __launch_bounds__(32)
void edge_message_kernel(
    const float* __restrict__ x, const int* __restrict__ ei,
    const float* __restrict__ ea,
    const v2f* __restrict__ Wm1p, const float* __restrict__ bm1,
    const float* __restrict__ gm,  const float* __restrict__ btm,
    const v2f* __restrict__ Wm2p, const float* __restrict__ bm2,
    const v2f* __restrict__ Wg1p, const float* __restrict__ bg1,
    const float* __restrict__ gg,  const float* __restrict__ btg,
    const float* __restrict__ Wg2, const float* __restrict__ bg2,
    float* __restrict__ aggr, int E_, int N_) {
  __shared__ float Alds[16 * 384];
  __shared__ float Hlds[16 * 128];
  __shared__ int   dsts[16];

  const int lane = threadIdx.x;
  const int e0   = blockIdx.x * 16;

  if (lane < 16) {
    int e = e0 + lane; if (e >= E_) e = E_ - 1;
    dsts[lane] = ei[E_ + e];                         // dst = edge_index[1]
  }
  __syncthreads();

  // Gather msg_input = [x[dst] | x[src] | edge_attr]  (16 x 384)
  for (int i = lane; i < 16 * 384; i += 32) {
    const int r = i / 384, c = i - r * 384;
    int e = e0 + r; if (e >= E_) e = E_ - 1;
    float val;
    if (c < 128)       val = x[(long)dsts[r] * 128 + c];
    else if (c < 256)  val = x[(long)ei[e]   * 128 + (c - 128)];   // src
    else               val = ea[(long)e      * 128 + (c - 256)];
    Alds[i] = val;
  }
  __syncthreads();

  // message MLP
  layer1_ln_silu<384>(Alds, Wm1p, bm1, gm, btm, Hlds);
  __syncthreads();
  v8f msg[8];
  layer2_128(Hlds, Wm2p, bm2, msg);
  __syncthreads();

  // gate MLP hidden layer (reuses Alds / Hlds)
  layer1_ln_silu<384>(Alds, Wg1p, bg1, gg, btg, Hlds);
  __syncthreads();

  // gate output layer: 128 -> 1, sigmoid. Lanes (l, l+16) share row l&15.
  const int hf = lane >> 4, col = lane & 15, row = lane & 15;
  float gdot = 0.f;
  for (int c = hf * 64; c < hf * 64 + 64; ++c)
    gdot += Hlds[row * 128 + c] * Wg2[c];
  gdot += __shfl_xor(gdot, 16);
  const float gate = 1.f / (1.f + __expf(-(gdot + bg2[0])));

  float gv[8];
#pragma unroll
  for (int v = 0; v < 8; ++v) gv[v] = __shfl(gate, v + 8 * hf);

  // scatter-add gate*msg into aggr[dst]  (global_atomic_add_f32)
#pragma unroll
  for (int n = 0; n < 8; ++n) {
#pragma unroll
    for (int v = 0; v < 8; ++v) {
      const int r = v + 8 * hf;
      const int e = e0 + r;
      if (e < E_) {
        float* p = &aggr[(long)dsts[r] * 128 + n * 16 + col];
        __hip_atomic_fetch_add(p, gv[v] * msg[n][v],
                               __ATOMIC_RELAXED, __HIP_MEMORY_SCOPE_AGENT);
      }
    }
  }
}

// ---------------------------------------------------------------------------
// Kernel 2: node update  x_new = MLP([x | aggr]) + x
// ---------------------------------------------------------------------------
__global__ __launch_bounds__(32)
void node_kernel(
    const float* __restrict__ x, const float* __restrict__ aggr,
    const v2f* __restrict__ W1p, const float* __restrict__ b1,
    const float* __restrict__ g,  const float* __restrict__ bt,
    const v2f* __restrict__ W2p, const float* __restrict__ b2,
    float* __restrict__ xnew, int N_) {
  __shared__ float Alds[16 * 256];
  __shared__ float Hlds[16 * 128];
  const int lane = threadIdx.x;
  const int t0   = blockIdx.x * 16;

  for (int i = lane; i < 16 * 256; i += 32) {
    const int r = i >> 8, c = i & 255;
    int nd = t0 + r; if (nd >= N_) nd = N_ - 1;
    Alds[i] = (c < 128) ? x[(long)nd * 128 + c]
                        : aggr[(long)nd * 128 + (c - 128)];
  }
  __syncthreads();

  layer1_ln_silu<256>(Alds, W1p, b1, g, bt, Hlds);
  __syncthreads();
  v8f out[8];
  layer2_128(Hlds, W2p, b2, out);

  const int hf = lane >> 4, col = lane & 15;
#pragma unroll
  for (int n = 0; n < 8; ++n) {
#pragma unroll
    for (int v = 0; v < 8; ++v) {
      const int nd = t0 + v + 8 * hf;
      if (nd < N_) {
        const long idx = (long)nd * 128 + n * 16 + col;
        xnew[idx] = out[n][v] + x[idx];          // residual
      }
    }
  }
}

// ---------------------------------------------------------------------------
// Kernel 3: edge update  e_new = MLP([edge_attr | x_new[src] | x_new[dst]]) + ea
// ---------------------------------------------------------------------------
__global__ __launch_bounds__(32)
void edge_update_kernel(
    const float* __restrict__ xnew, const int* __restrict__ ei,
    const float* __restrict__ ea,
    const v2f* __restrict__ W1p, const float* __restrict__ b1,
    const float* __restrict__ g,  const float* __restrict__ bt,
    const v2f* __restrict__ W2p, const float* __restrict__ b2,
    float* __restrict__ eout, int E_) {
  __shared__ float Alds[16 * 384];
  __shared__ float Hlds[16 * 128];
  const int lane = threadIdx.x;
  const int e0   = blockIdx.x * 16;

  for (int i = lane; i < 16 * 384; i += 32) {
    const int r = i / 384, c = i - r * 384;
    int e = e0 + r; if (e >= E_) e = E_ - 1;
    float val;
    if (c < 128)       val = ea[(long)e * 128 + c];
    else if (c < 256)  val = xnew[(long)ei[e]      * 128 + (c - 128)]; // x_new[src]
    else               val = xnew[(long)ei[E_ + e] * 128 + (c - 256)]; // x_new[dst]
    Alds[i] = val;
  }
  __syncthreads();

  layer1_ln_silu<384>(Alds, W1p, b1, g, bt, Hlds);
  __syncthreads();
  v8f out[8];
  layer2_128(Hlds, W2p, b2, out);

  const int hf = lane >> 4, col = lane & 15;
#pragma unroll
  for (int n = 0; n < 8; ++n) {
#pragma unroll
    for (int v = 0; v < 8; ++v) {
      const int e = e0 + v + 8 * hf;
      if (e < E_) {
        const long idx = (long)e * 128 + n * 16 + col;
        eout[idx] = out[n][v] + ea[idx];         // residual
      }
    }
  }
}

// ---------------------------------------------------------------------------
// Host launcher
// ---------------------------------------------------------------------------
extern "C" void kernel_launch(void* const* d_in, const int* in_sizes, int n_in,
                              void* d_out, int out_size, void* d_ws, size_t ws_size,
                              hipStream_t stream) {
  // setup_inputs() insertion order: x, edge_index, edge_attr,
  // params{message,gate,node,edge}, each MLP: W1, b1, ln_g, ln_b, W2, b2
  const float* x   = (const float*)d_in[0];
  const int*   ei  = (const int*)d_in[1];
  const float* ea  = (const float*)d_in[2];

  const float* Wm1 = (const float*)d_in[3];  const float* bm1 = (const float*)d_in[4];
  const float* gm  = (const float*)d_in[5];  const float* btm = (const float*)d_in[6];
  const float* Wm2 = (const float*)d_in[7];  const float* bm2 = (const float*)d_in[8];

  const float* Wg1 = (const float*)d_in[9];  const float* bg1 = (const float*)d_in[10];
  const float* gg  = (const float*)d_in[11]; const float* btg = (const float*)d_in[12];
  const float* Wg2 = (const float*)d_in[13]; const float* bg2 = (const float*)d_in[14];

  const float* Wn1 = (const float*)d_in[15]; const float* bn1 = (const float*)d_in[16];
  const float* gn  = (const float*)d_in[17]; const float* btn = (const float*)d_in[18];
  const float* Wn2 = (const float*)d_in[19]; const float* bn2 = (const float*)d_in[20];

  const float* We1 = (const float*)d_in[21]; const float* be1 = (const float*)d_in[22];
  const float* ge  = (const float*)d_in[23]; const float* bte = (const float*)d_in[24];
  const float* We2 = (const float*)d_in[25]; const float* be2 = (const float*)d_in[26];

  const int N_ = in_sizes[0] / 128;   // 20000
  const int E_ = in_sizes[1] / 2;     // 640000

  // ---- workspace layout (floats) ----
  float* ws   = (float*)d_ws;
  float* aggr = ws;                          size_t off = (size_t)N_ * 128;
  float* Wm1p = ws + off;                    off += 384 * 128;
  float* Wg1p = ws + off;                    off += 384 * 128;
  float* Wn1p = ws + off;                    off += 256 * 128;
  float* We1p = ws + off;                    off += 384 * 128;
  float* Wm2p = ws + off;                    off += 128 * 128;
  float* Wn2p = ws + off;                    off += 128 * 128;
  float* We2p = ws + off;                    off += 128 * 128;

  float* xnew = (float*)d_out;                      // N x 128
  float* enew = (float*)d_out + (size_t)N_ * 128;   // E x 128

  // ---- pack weights into WMMA fragment order ----
  {
    const int t384 = 384 * 64, t256 = 256 * 64, t128 = 128 * 64; // (KIN/4)*8*32
    pack_w_kernel<<<(t384 + 255) / 256, 256, 0, stream>>>(Wm1, Wm1p, 384);
    pack_w_kernel<<<(t384 + 255) / 256, 256, 0, stream>>>(Wg1, Wg1p, 384);
    pack_w_kernel<<<(t256 + 255) / 256, 256, 0, stream>>>(Wn1, Wn1p, 256);
    pack_w_kernel<<<(t384 + 255) / 256, 256, 0, stream>>>(We1, We1p, 384);
    pack_w_kernel<<<(t128 + 255) / 256, 256, 0, stream>>>(Wm2, Wm2p, 128);
    pack_w_kernel<<<(t128 + 255) / 256, 256, 0, stream>>>(Wn2, Wn2p, 128);
    pack_w_kernel<<<(t128 + 255) / 256, 256, 0, stream>>>(We2, We2p, 128);
  }

  const long an = (long)N_ * 128;
  zero_f32<<<(unsigned)((an + 255) / 256), 256, 0, stream>>>(aggr, an);

  const int etiles = (E_ + 15) / 16;
  const int ntiles = (N_ + 15) / 16;

  edge_message_kernel<<<etiles, 32, 0, stream>>>(
      x, ei, ea, (const v2f*)Wm1p, bm1, gm, btm, (const v2f*)Wm2p, bm2,
      (const v2f*)Wg1p, bg1, gg, btg, Wg2, bg2, aggr, E_, N_);

  node_kernel<<<ntiles, 32, 0, stream>>>(
      x, aggr, (const v2f*)Wn1p, bn1, gn, btn, (const v2f*)Wn2p, bn2, xnew, N_);

  edge_update_kernel<<<etiles, 32, 0, stream>>>(
      xnew, ei, ea, (const v2f*)We1p, be1, ge, bte, (const v2f*)We2p, be2, enew, E_);
}